// HebbyRNN_52201032516012
// MI455X (gfx1250) — compile-verified
//
#include <hip/hip_runtime.h>
#include <hip/hip_bf16.h>

typedef __attribute__((ext_vector_type(2))) float v2f;
typedef __attribute__((ext_vector_type(8))) float v8f;

#define B_   32
#define IN_  128
#define H_   1024
#define OUT_ 128

// Batched per-sample GEMV:  out[b,o] = act( sum_k W[b,o,k] * in[b,k] + bias[o] )
// in = concat(in0[b, 0:in0len], in1[b, 0:K-in0len])
// grid: (Hout/32, B), block: 256 threads = 8 waves.
// Each block: 2 tiles of 16 output rows; 4 waves per tile split K into quarters.
// Wave-level math: V_WMMA_F32_16X16X4_F32 with A = 16x4 f32 weight tile streamed
// from HBM (one float2 per lane per WMMA step), B = x in column 0 only, so D
// column 0 (lanes 0 and 16) accumulates the 16 dot products.
// Inner loop manually unrolled 8x (K-step 32) so 8 global_load_b64 are in
// flight per wave before the WMMA batch consumes them.
__global__ __launch_bounds__(256) void hebb_gemv_wmma(
    const float* __restrict__ W, const float* __restrict__ bias,
    const float* __restrict__ in0, const float* __restrict__ in1,
    int in0len, int K, int Hout, float* __restrict__ out, int doTanh)
{
  __shared__ float s_in[1152];
  __shared__ float s_red[8][16];

  const int tid = threadIdx.x;
  const int b   = blockIdx.y;

  // Stage the per-sample input vector in LDS (reused by all 8 waves).
  for (int i = tid; i < K; i += 256) {
    float v;
    if (i < in0len) v = in0[(size_t)b * in0len + i];
    else            v = in1[(size_t)b * (size_t)(K - in0len) + (i - in0len)];
    s_in[i] = v;
  }
  __syncthreads();

  const int wave      = tid >> 5;
  const int lane      = tid & 31;
  const int tileInBlk = wave >> 2;   // 0..1 : which 16-row tile
  const int ksplit    = wave & 3;    // 0..3 : which K quarter
  const int r         = lane & 15;   // A-matrix row within tile
  const int hi        = lane >> 4;   // 0: K={0,1}, 1: K={2,3}
  const int rbase     = blockIdx.x * 32 + tileInBlk * 16;
  const int kq        = K >> 2;      // 288 or 256, both % 32 == 0
  const int k0        = ksplit * kq;
  const int k1        = k0 + kq;

  // Lane's weight stream: row (rbase+r), starting at column 2*hi.
  const float* Wp = W + ((size_t)b * Hout + (rbase + r)) * (size_t)K + 2 * hi;
  // Prefetch stream: per lane-half, one 64B line per 32-column group.
  const float* Pp = W + ((size_t)b * Hout + (rbase + r)) * (size_t)K + 16 * hi;

  v8f c = {};
  for (int kb = k0; kb < k1; kb += 32) {
    // Pull the group ~2 groups ahead into cache (emits global_prefetch_b8).
    __builtin_prefetch(Pp + kb + 64, 0, 1);

    // 8 A fragments: lane holds W[row, kb+4j+2*hi .. +1]
    v2f a0 = *(const v2f*)(Wp + kb);
    v2f a1 = *(const v2f*)(Wp + kb + 4);
    v2f a2 = *(const v2f*)(Wp + kb + 8);
    v2f a3 = *(const v2f*)(Wp + kb + 12);
    v2f a4 = *(const v2f*)(Wp + kb + 16);
    v2f a5 = *(const v2f*)(Wp + kb + 20);
    v2f a6 = *(const v2f*)(Wp + kb + 24);
    v2f a7 = *(const v2f*)(Wp + kb + 28);

    // B fragments: column N=0 carries x; lanes with (lane%16)==0 are N=0.
    const int kx = kb + 2 * hi;
    const bool n0 = (r == 0);
    v2f b0, b1, b2, b3, b4, b5, b6, b7;
    b0.x = n0 ? s_in[kx +  0] : 0.0f;  b0.y = n0 ? s_in[kx +  1] : 0.0f;
    b1.x = n0 ? s_in[kx +  4] : 0.0f;  b1.y = n0 ? s_in[kx +  5] : 0.0f;
    b2.x = n0 ? s_in[kx +  8] : 0.0f;  b2.y = n0 ? s_in[kx +  9] : 0.0f;
    b3.x = n0 ? s_in[kx + 12] : 0.0f;  b3.y = n0 ? s_in[kx + 13] : 0.0f;
    b4.x = n0 ? s_in[kx + 16] : 0.0f;  b4.y = n0 ? s_in[kx + 17] : 0.0f;
    b5.x = n0 ? s_in[kx + 20] : 0.0f;  b5.y = n0 ? s_in[kx + 21] : 0.0f;
    b6.x = n0 ? s_in[kx + 24] : 0.0f;  b6.y = n0 ? s_in[kx + 25] : 0.0f;
    b7.x = n0 ? s_in[kx + 28] : 0.0f;  b7.y = n0 ? s_in[kx + 29] : 0.0f;

    c = __builtin_amdgcn_wmma_f32_16x16x4_f32(false, a0, false, b0, (short)0, c, false, false);
    c = __builtin_amdgcn_wmma_f32_16x16x4_f32(false, a1, false, b1, (short)0, c, false, false);
    c = __builtin_amdgcn_wmma_f32_16x16x4_f32(false, a2, false, b2, (short)0, c, false, false);
    c = __builtin_amdgcn_wmma_f32_16x16x4_f32(false, a3, false, b3, (short)0, c, false, false);
    c = __builtin_amdgcn_wmma_f32_16x16x4_f32(false, a4, false, b4, (short)0, c, false, false);
    c = __builtin_amdgcn_wmma_f32_16x16x4_f32(false, a5, false, b5, (short)0, c, false, false);
    c = __builtin_amdgcn_wmma_f32_16x16x4_f32(false, a6, false, b6, (short)0, c, false, false);
    c = __builtin_amdgcn_wmma_f32_16x16x4_f32(false, a7, false, b7, (short)0, c, false, false);
  }

  // D column 0: lane 0 -> rows rbase+0..7 in VGPRs 0..7, lane 16 -> rows rbase+8..15.
  if (lane == 0) {
    #pragma unroll
    for (int j = 0; j < 8; ++j) s_red[wave][j] = c[j];
  } else if (lane == 16) {
    #pragma unroll
    for (int j = 0; j < 8; ++j) s_red[wave][8 + j] = c[j];
  }
  __syncthreads();

  // Reduce the 4 K-split partials per tile; bias + activation; store.
  if (tid < 32) {
    const int tile = tid >> 4;
    const int rl   = tid & 15;
    float s = s_red[tile * 4 + 0][rl] + s_red[tile * 4 + 1][rl]
            + s_red[tile * 4 + 2][rl] + s_red[tile * 4 + 3][rl];
    const int orow = blockIdx.x * 32 + tid;
    s += bias[orow];
    if (doTanh) s = tanhf(s);
    out[(size_t)b * Hout + orow] = s;
  }
}

// Row-wise log_softmax over OUT_=128 logits. grid: B, block: 128.
__global__ __launch_bounds__(128) void hebb_logsoftmax(
    const float* __restrict__ logits, float* __restrict__ out)
{
  __shared__ float red[128];
  const int b = blockIdx.x;
  const int t = threadIdx.x;
  const float v = logits[b * OUT_ + t];

  red[t] = v;
  __syncthreads();
  for (int s = 64; s > 0; s >>= 1) {
    if (t < s) red[t] = fmaxf(red[t], red[t + s]);
    __syncthreads();
  }
  const float m = red[0];
  __syncthreads();

  red[t] = expf(v - m);
  __syncthreads();
  for (int s = 64; s > 0; s >>= 1) {
    if (t < s) red[t] += red[t + s];
    __syncthreads();
  }
  out[b * OUT_ + t] = v - m - logf(red[0]);
}

extern "C" void kernel_launch(void* const* d_in, const int* in_sizes, int n_in,
                              void* d_out, int out_size, void* d_ws, size_t ws_size,
                              hipStream_t stream) {
  const float* x  = (const float*)d_in[0];
  const float* h  = (const float*)d_in[1];
  const float* W0 = (const float*)d_in[2];
  const float* b0 = (const float*)d_in[3];
  const float* W1 = (const float*)d_in[4];
  const float* b1 = (const float*)d_in[5];
  const float* W2 = (const float*)d_in[6];
  const float* b2 = (const float*)d_in[7];
  const float* Wh = (const float*)d_in[8];
  const float* bh = (const float*)d_in[9];
  const float* Wo = (const float*)d_in[10];
  const float* bo = (const float*)d_in[11];

  float* out = (float*)d_out;
  float* ws  = (float*)d_ws;

  float* c1 = ws;                 // [B, H]
  float* c2 = ws + (size_t)B_ * H_;
  float* c3 = ws + (size_t)2 * B_ * H_;
  float* lg = ws + (size_t)3 * B_ * H_;   // [B, OUT]

  float* outLogits = out;                     // log_softmax output [B, OUT]
  float* outHidden = out + (size_t)B_ * OUT_; // new_hidden        [B, H]

  const dim3 blk(256);
  // layer 0: (IN+H) -> H, input = concat(x, hidden)
  hebb_gemv_wmma<<<dim3(H_ / 32, B_), blk, 0, stream>>>(
      W0, b0, x, h, IN_, IN_ + H_, H_, c1, 1);
  // layer 1
  hebb_gemv_wmma<<<dim3(H_ / 32, B_), blk, 0, stream>>>(
      W1, b1, c1, c1, H_, H_, H_, c2, 1);
  // layer 2
  hebb_gemv_wmma<<<dim3(H_ / 32, B_), blk, 0, stream>>>(
      W2, b2, c2, c2, H_, H_, H_, c3, 1);
  // i2h head -> new_hidden (second output)
  hebb_gemv_wmma<<<dim3(H_ / 32, B_), blk, 0, stream>>>(
      Wh, bh, c3, c3, H_, H_, H_, outHidden, 1);
  // i2o head -> logits (no activation)
  hebb_gemv_wmma<<<dim3(OUT_ / 32, B_), blk, 0, stream>>>(
      Wo, bo, c3, c3, H_, H_, OUT_, lg, 0);
  // log_softmax -> first output
  hebb_logsoftmax<<<dim3(B_), dim3(128), 0, stream>>>(lg, outLogits);
}